// Forward_48524540510237
// MI455X (gfx1250) — compile-verified
//
#include <hip/hip_runtime.h>

// 3D leapfrog wave step with sponge damping, 320^3 fp32.
// Memory-bound (~0.92 GB traffic, ~0.65 GFLOP) -> HBM roofline target.
// Fused single pass; LDS ring buffer of s = p_cur + source_cur (3 z-planes
// with 1-wide halo); CDNA5 async global->LDS staging (GLOBAL_LOAD_ASYNC_TO_LDS,
// ASYNCcnt) software-pipelined one plane ahead of compute.

#define Dn 320
#define Hn 320
#define Wn 320
#define DT_F 1e-3f

#define TX 64               // tile x (contiguous)
#define TY 16               // tile y
#define CZ 32               // z chunk per block
#define HX (TX + 2)         // 66  halo tile x
#define HY (TY + 2)         // 18  halo tile y
#define NT (HX * HY)        // 1188 halo tile elems
#define NTHREADS (TX * TY)  // 1024 threads = 32 wave32

#if __has_builtin(__builtin_amdgcn_global_load_async_to_lds_b32)
#define HAS_ASYNC_LDS 1
#else
#define HAS_ASYNC_LDS 0
#endif

typedef __attribute__((address_space(1))) int gint_t;   // global (AS1)
typedef __attribute__((address_space(3))) int lint_t;   // LDS (AS3)

__device__ __forceinline__ void async_cp_b32(const float* g, float* l) {
#if HAS_ASYNC_LDS
    __builtin_amdgcn_global_load_async_to_lds_b32((gint_t*)g, (lint_t*)l, 0, 0);
#else
    *l = *g;
#endif
}

__device__ __forceinline__ void wait_async_all() {
#if __has_builtin(__builtin_amdgcn_s_wait_asynccnt)
    __builtin_amdgcn_s_wait_asynccnt(0);
#elif HAS_ASYNC_LDS
    asm volatile("s_wait_asynccnt 0" ::: "memory");
#endif
}

__global__ __launch_bounds__(NTHREADS) void sponge_step_kernel(
    const float* __restrict__ source_cur,
    const float* __restrict__ p_cur,
    const float* __restrict__ p_pre,
    const float* __restrict__ sigma,
    const float* __restrict__ a,
    float* __restrict__ out_pnew,
    float* __restrict__ out_pcur)
{
    __shared__ float sP[3][NT];   // ring buffer of s-planes (z-1, z, z+1)
    __shared__ float stgP[NT];    // staging: raw p_cur plane (tile + halo)
    __shared__ float stgS[NT];    // staging: raw source plane (tile + halo)

    const int tid = threadIdx.y * TX + threadIdx.x;
    const int x0 = blockIdx.x * TX;
    const int y0 = blockIdx.y * TY;
    const int z0 = blockIdx.z * CZ;
    const int z1 = z0 + CZ;

    // Each thread owns at most two halo-tile slots; precompute them once.
    const int i0  = tid;
    const int gy0 = y0 - 1 + (i0 / HX);
    const int gx0 = x0 - 1 + (i0 % HX);
    const bool ok0 = (gy0 >= 0) & (gy0 < Hn) & (gx0 >= 0) & (gx0 < Wn);
    const size_t off0 = ok0 ? ((size_t)gy0 * Wn + (size_t)gx0) : 0;

    const int i1  = tid + NTHREADS;
    const bool has1 = (i1 < NT);                 // only tid < NT-NTHREADS
    const int gy1 = y0 - 1 + (i1 / HX);
    const int gx1 = x0 - 1 + (i1 % HX);
    const bool ok1 = (gy1 >= 0) & (gy1 < Hn) & (gx1 >= 0) & (gx1 < Wn);
    const size_t off1 = ok1 ? ((size_t)gy1 * Wn + (size_t)gx1) : 0;

    const size_t planeSz = (size_t)Hn * Wn;

    // Issue async copies of raw plane zp (tile + halo) into staging LDS.
    auto issue_plane = [&](int zp) {
        const float* pz = p_cur + (size_t)zp * planeSz;
        const float* sz = source_cur + (size_t)zp * planeSz;
        if (ok0) {
            async_cp_b32(pz + off0, &stgP[i0]);
            async_cp_b32(sz + off0, &stgS[i0]);
        }
        if (has1 & ok1) {
            async_cp_b32(pz + off1, &stgP[i1]);
            async_cp_b32(sz + off1, &stgS[i1]);
        }
    };

    // Build s-plane in ring slot from staged raw data (zeros when z OOB).
    auto build_plane = [&](int slot, int zp) {
        const bool zv = (zp >= 0) & (zp < Dn);
        sP[slot][i0] = zv ? (stgP[i0] + stgS[i0]) : 0.0f;
        if (has1) sP[slot][i1] = zv ? (stgP[i1] + stgS[i1]) : 0.0f;
    };

    // Start the first DMA immediately (disjoint LDS slots from the zeroing).
    if (z0 - 1 >= 0) issue_plane(z0 - 1);

    // xy-out-of-range staging slots are never written by the async copies:
    // zero them exactly once (implements conv zero padding in x/y).
    if (!ok0)        { stgP[i0] = 0.0f; stgS[i0] = 0.0f; }
    if (has1 & !ok1) { stgP[i1] = 0.0f; stgS[i1] = 0.0f; }

    wait_async_all();
    __syncthreads();
    build_plane(0, z0 - 1);
    __syncthreads();

    issue_plane(z0);
    wait_async_all();
    __syncthreads();
    build_plane(1, z0);
    __syncthreads();

    issue_plane(z0 + 1);   // z0+1 <= 289 < Dn always

    const int ci = (threadIdx.y + 1) * HX + (threadIdx.x + 1);
    const int x = x0 + threadIdx.x;
    const int y = y0 + threadIdx.y;
    const size_t base2d = (size_t)y * Wn + (size_t)x;

    for (int z = z0; z < z1; ++z) {
        const int r    = (z - z0) % 3;
        const int prev = r;
        const int cur  = (r + 1) % 3;
        const int next = (r + 2) % 3;

        // staging now holds raw plane z+1
        wait_async_all();
        __syncthreads();
        build_plane(next, z + 1);
        __syncthreads();   // staging consumed; sP[next] visible to all waves

        // prefetch raw plane z+2 for the next iteration (overlaps compute)
        if ((z + 1 < z1) & (z + 2 < Dn)) issue_plane(z + 2);

        // ---- compute plane z ----
        const float sC  = sP[cur][ci];
        const float lap = sP[prev][ci] + sP[next][ci]
                        + sP[cur][ci - 1]  + sP[cur][ci + 1]
                        + sP[cur][ci - HX] + sP[cur][ci + HX]
                        - 6.0f * sC;

        const size_t gi = (size_t)z * planeSz + base2d;
        const float sg = __builtin_nontemporal_load(sigma + gi);
        const float pp = __builtin_nontemporal_load(p_pre + gi);
        const float av = __builtin_nontemporal_load(a + gi);

        const float hh = 0.5f * DT_F * sg;
        const float pn = (2.0f * sC - (1.0f - hh) * pp + av * lap)
                       / (1.0f + hh);

        __builtin_nontemporal_store(pn, out_pnew + gi);
        __builtin_nontemporal_store(sC, out_pcur + gi);
    }
}

extern "C" void kernel_launch(void* const* d_in, const int* in_sizes, int n_in,
                              void* d_out, int out_size, void* d_ws, size_t ws_size,
                              hipStream_t stream) {
    (void)in_sizes; (void)n_in; (void)out_size; (void)d_ws; (void)ws_size;

    const float* source_cur = (const float*)d_in[0];
    const float* p_cur      = (const float*)d_in[1];
    const float* p_pre      = (const float*)d_in[2];
    const float* sigma      = (const float*)d_in[3];
    const float* a          = (const float*)d_in[4];

    const size_t N = (size_t)Dn * Hn * Wn;
    float* out_pnew = (float*)d_out;        // first return: p_new
    float* out_pcur = (float*)d_out + N;    // second return: p_cur + source

    dim3 grid(Wn / TX, Hn / TY, Dn / CZ);   // (5, 20, 10) = 1000 blocks
    dim3 block(TX, TY, 1);                  // 1024 threads = 32 wave32

    hipLaunchKernelGGL(sponge_step_kernel, grid, block, 0, stream,
                       source_cur, p_cur, p_pre, sigma, a, out_pnew, out_pcur);
}